// GNN_19404662243922
// MI455X (gfx1250) — compile-verified
//
#include <hip/hip_runtime.h>
#include <hip/hip_bf16.h>

typedef __attribute__((ext_vector_type(2))) float v2f;
typedef __attribute__((ext_vector_type(8))) float v8f;

// ---------------------------------------------------------------- utilities

__global__ void k_zero(float* __restrict__ p, long long n) {
  long long i = (long long)blockIdx.x * blockDim.x + threadIdx.x;
  if (i < n) p[i] = 0.0f;
}

// degree histogram over edge destinations (self-loop added in k_dinv)
__global__ void k_deg(const long long* __restrict__ dst, float* __restrict__ deg, int E) {
  int e = blockIdx.x * blockDim.x + threadIdx.x;
  if (e < E) atomicAdd(&deg[(int)dst[e]], 1.0f);
}

// dinv = rsqrt(deg_edges + 1)   (+1 = self loop; always > 0)
__global__ void k_dinv(float* __restrict__ deg, int n) {
  int i = blockIdx.x * blockDim.x + threadIdx.x;
  if (i < n) deg[i] = rsqrtf(deg[i] + 1.0f);
}

// ---------------------------------------------------------------- WMMA GEMM
// One wave computes a 16-row tile of  out[M,NOUT] = A[M,K] @ W[K,NOUT]
// using V_WMMA_F32_16X16X4_F32 (exact fp32).
// A layout : lane = M (0..15, both halves), VGPR = K%2, lane-half = K/2
// B layout : lane = N, VGPR = K%2, lane-half = K/2
// C/D      : VGPR r -> M = r + 8*half, lane -> N

template <int K, int NOUT, bool RELU, bool BIAS>
__global__ void k_gemm(const float* __restrict__ A, const float* __restrict__ W,
                       const float* __restrict__ bias, float* __restrict__ out,
                       int ntiles) {
  const int wave = threadIdx.x >> 5;
  const int tile = blockIdx.x * (blockDim.x >> 5) + wave;
  if (tile >= ntiles) return;                    // wave-uniform: EXEC stays full
  const int lane = threadIdx.x & 31;
  const int hf   = lane >> 4;                    // lane half (0/1)
  const int lr   = lane & 15;

  constexpr int NT = NOUT / 16;                  // 16-wide N tiles
  v8f c[NT] = {};

  const float* arow = A + (size_t)(tile * 16 + lr) * K;

#pragma unroll
  for (int kc = 0; kc < K / 4; ++kc) {
    const int kk = kc * 4 + 2 * hf;              // this lane-half's K pair
    v2f a = *(const v2f*)(arow + kk);            // A[m][kk], A[m][kk+1]
#pragma unroll
    for (int t = 0; t < NT; ++t) {
      v2f b;
      b.x = W[(size_t)kk * NOUT + t * 16 + lr];        // B[kk  ][n]
      b.y = W[(size_t)(kk + 1) * NOUT + t * 16 + lr];  // B[kk+1][n]
      c[t] = __builtin_amdgcn_wmma_f32_16x16x4_f32(
          /*neg_a=*/false, a, /*neg_b=*/false, b,
          /*c_mod=*/(short)0, c[t], /*reuse_a=*/false, /*reuse_b=*/false);
    }
  }

#pragma unroll
  for (int t = 0; t < NT; ++t) {
    const float bv = BIAS ? bias[t * 16 + lr] : 0.0f;
#pragma unroll
    for (int r = 0; r < 8; ++r) {
      float v = c[t][r] + bv;
      if (RELU) v = fmaxf(v, 0.0f);
      out[(size_t)(tile * 16 + r + 8 * hf) * NOUT + t * 16 + lr] = v;
    }
  }
}

// ---------------------------------------------------------------- aggregation
// agg[dst] += h[src] * dinv[src]*dinv[dst]   (one thread per edge)
template <int F>
__global__ void k_aggregate(const long long* __restrict__ ei, const float* __restrict__ h,
                            const float* __restrict__ dinv, float* __restrict__ agg, int E) {
  int e = blockIdx.x * blockDim.x + threadIdx.x;
  if (e >= E) return;
  const int s = (int)ei[e];
  const int d = (int)ei[(size_t)E + e];
  const float norm = dinv[s] * dinv[d];
  const float4* hs = (const float4*)(h + (size_t)s * F);
  float* ad = agg + (size_t)d * F;
#pragma unroll
  for (int q = 0; q < F / 4; ++q) {
    float4 v = hs[q];
    atomicAdd(&ad[q * 4 + 0], v.x * norm);
    atomicAdd(&ad[q * 4 + 1], v.y * norm);
    atomicAdd(&ad[q * 4 + 2], v.z * norm);
    atomicAdd(&ad[q * 4 + 3], v.w * norm);
  }
}

// out = relu(agg + h*dinv^2 + b)   (self-loop term, bias, activation)
template <int F, bool RELU>
__global__ void k_finalize(const float* __restrict__ agg, const float* __restrict__ h,
                           const float* __restrict__ dinv, const float* __restrict__ bias,
                           float* __restrict__ out, int n) {
  int i = blockIdx.x * blockDim.x + threadIdx.x;
  if (i >= n * F) return;
  const int node = i / F;
  const int f = i % F;
  const float di = dinv[node];
  float v = agg[i] + h[i] * di * di + bias[f];
  if (RELU) v = fmaxf(v, 0.0f);
  out[i] = v;
}

// ---------------------------------------------------------------- launcher

extern "C" void kernel_launch(void* const* d_in, const int* in_sizes, int n_in,
                              void* d_out, int out_size, void* d_ws, size_t ws_size,
                              hipStream_t stream) {
  const float* x   = (const float*)d_in[0];
  const float* W1  = (const float*)d_in[1];
  const float* b1  = (const float*)d_in[2];
  const float* W2  = (const float*)d_in[3];
  const float* b2  = (const float*)d_in[4];
  const float* Wf1 = (const float*)d_in[5];
  const float* bf1 = (const float*)d_in[6];
  const float* Wf2 = (const float*)d_in[7];
  const float* bf2 = (const float*)d_in[8];
  const float* Wf3 = (const float*)d_in[9];
  const float* bf3 = (const float*)d_in[10];
  const long long* ei = (const long long*)d_in[11];
  float* out = (float*)d_out;

  const int N = in_sizes[0] / 128;   // 100000
  const int E = in_sizes[11] / 2;    // 6400000

  // workspace layout (floats)
  float* ws   = (float*)d_ws;
  float* dinv = ws;                          //  1*N (deg -> dinv in place)
  float* h1   = dinv + (size_t)N;            // 16*N
  float* a1   = h1 + (size_t)16 * N;         // 16*N (agg1 -> a1 in place)
  float* h2   = a1 + (size_t)16 * N;         // 32*N
  float* a2   = h2 + (size_t)32 * N;         // 32*N (agg2 -> a2 in place)
  float* z1   = a2 + (size_t)32 * N;         // 64*N
  float* z2   = z1 + (size_t)64 * N;         // 32*N

  const int B = 256;
  const int ntiles = N / 16;                         // 6250
  const int gemmBlocks = (ntiles + 7) / 8;           // 8 waves / block
  const int edgeBlocks = (E + B - 1) / B;

  // zero deg + both agg buffers (covers ws[0 .. 97N))
  {
    long long zn = (long long)(1 + 16 + 16 + 32 + 32) * N;
    k_zero<<<(int)((zn + B - 1) / B), B, 0, stream>>>(ws, zn);
  }

  // normalization
  k_deg<<<edgeBlocks, B, 0, stream>>>(ei + E, dinv, E);
  k_dinv<<<(N + B - 1) / B, B, 0, stream>>>(dinv, N);

  // layer 1: h1 = x@W1 ; aggregate ; a1 = relu(agg + h1*dinv^2 + b1)
  k_gemm<128, 16, false, false><<<gemmBlocks, B, 0, stream>>>(x, W1, nullptr, h1, ntiles);
  k_aggregate<16><<<edgeBlocks, B, 0, stream>>>(ei, h1, dinv, a1, E);
  k_finalize<16, true><<<(N * 16 + B - 1) / B, B, 0, stream>>>(a1, h1, dinv, b1, a1, N);

  // layer 2: h2 = a1@W2 ; aggregate ; a2 = relu(agg + h2*dinv^2 + b2)
  k_gemm<16, 32, false, false><<<gemmBlocks, B, 0, stream>>>(a1, W2, nullptr, h2, ntiles);
  k_aggregate<32><<<edgeBlocks, B, 0, stream>>>(ei, h2, dinv, a2, E);
  k_finalize<32, true><<<(N * 32 + B - 1) / B, B, 0, stream>>>(a2, h2, dinv, b2, a2, N);

  // MLP head: relu(a2@Wf1+bf1) -> relu(z1@Wf2+bf2) -> z2@Wf3+bf3
  k_gemm<32, 64, true,  true><<<gemmBlocks, B, 0, stream>>>(a2, Wf1, bf1, z1, ntiles);
  k_gemm<64, 32, true,  true><<<gemmBlocks, B, 0, stream>>>(z1, Wf2, bf2, z2, ntiles);
  k_gemm<32, 16, false, true><<<gemmBlocks, B, 0, stream>>>(z2, Wf3, bf3, out, ntiles);
}